// HCHA_9337258901911
// MI455X (gfx1250) — compile-verified
//
#include <hip/hip_runtime.h>
#include <hip/hip_bf16.h>
#include <math.h>

// ---------------------------------------------------------------------------
// HypergraphConv x3 (HCHA, use_attention=False) for MI455X / gfx1250.
//   per layer: XW = X @ W   (fp32 WMMA 16x16x4)
//              m   = B^-1 * segsum_hedge(XW[node_idx])      (float4 atomics)
//              out = elu(D^-1 * segsum_node(m[hedge_idx]) + b)
// Degrees computed once (indices shared across layers).
// ---------------------------------------------------------------------------

typedef __attribute__((ext_vector_type(2))) float v2f;
typedef __attribute__((ext_vector_type(8))) float v8f;

// ---------------- GEMM: C[N,128] = A[N,128] @ W[128,128], fp32 WMMA -------
__global__ __launch_bounds__(256) void hgc_gemm_wmma(
    const float* __restrict__ A, const float* __restrict__ W,
    float* __restrict__ C, int N) {
  const int lane = threadIdx.x & 31;
  const int wave = threadIdx.x >> 5;       // 8 waves -> 8 column tiles of 16
  const int row0 = blockIdx.x << 4;        // 16 rows per block
  const int col0 = wave << 4;
  const int m    = lane & 15;
  const int kh2  = (lane >> 4) << 1;       // 0 or 2 (K sub-offset per half-wave)

  int row = row0 + m;
  if (row > N - 1) row = N - 1;            // clamp loads; EXEC stays all-ones

  const float* a_base = A + (size_t)row * 128 + kh2;      // A[row][k0+kh2]
  const float* b_base = W + (size_t)kh2 * 128 + col0 + m; // W[k0+kh2][col]

  v8f acc = {0.f, 0.f, 0.f, 0.f, 0.f, 0.f, 0.f, 0.f};
#pragma unroll
  for (int k0 = 0; k0 < 128; k0 += 4) {
    v2f a = *(const v2f*)(a_base + k0);    // 8B aligned: kh2,k0 even
    v2f b;
    b.x = b_base[(size_t)k0 * 128];
    b.y = b_base[(size_t)k0 * 128 + 128];
    acc = __builtin_amdgcn_wmma_f32_16x16x4_f32(
        /*neg_a=*/false, a, /*neg_b=*/false, b,
        /*c_mod=*/(short)0, acc, /*reuse_a=*/false, /*reuse_b=*/false);
  }

  // C/D layout: VGPR i -> row row0+i (lanes 0-15), row0+8+i (lanes 16-31)
  const int rbase = row0 + ((lane >> 4) << 3);
  const int c     = col0 + m;
#pragma unroll
  for (int i = 0; i < 8; ++i) {
    int r = rbase + i;
    if (r < N) C[(size_t)r * 128 + c] = acc[i];
  }
}

// ---------------- degree histogram + inversion -----------------------------
__global__ __launch_bounds__(256) void hgc_degrees(
    const int* __restrict__ ni, const int* __restrict__ hi,
    float* __restrict__ dn, float* __restrict__ de, int E) {
  int i = blockIdx.x * blockDim.x + threadIdx.x;
  if (i >= E) return;
  atomicAdd(dn + ni[i], 1.0f);
  atomicAdd(de + hi[i], 1.0f);
}

__global__ __launch_bounds__(256) void hgc_invert(float* __restrict__ d, int n) {
  int i = blockIdx.x * blockDim.x + threadIdx.x;
  if (i >= n) return;
  float v = d[i];
  d[i] = (v > 0.0f) ? (1.0f / v) : 0.0f;
}

// ---------------- scatter-add: one wave per incidence, float4 lanes --------
__global__ __launch_bounds__(256) void hgc_scatter_add(
    const float4* __restrict__ src, const int* __restrict__ gidx,
    const int* __restrict__ sidx, float* __restrict__ dst, int E) {
  int inc = (blockIdx.x << 3) + (threadIdx.x >> 5);   // 8 incidences / block
  if (inc >= E) return;
  const int lane = threadIdx.x & 31;
  const int g = gidx[inc];                 // wave-uniform
  const int s = sidx[inc];                 // wave-uniform
  float4 v = src[((size_t)g << 5) + lane]; // 128 floats = 32 lanes x float4
  float* p = dst + ((size_t)s << 7) + (lane << 2);
  atomicAdd(p + 0, v.x);
  atomicAdd(p + 1, v.y);
  atomicAdd(p + 2, v.z);
  atomicAdd(p + 3, v.w);
}

// ---------------- row scaling: d[r][:] *= scale[r] -------------------------
__global__ __launch_bounds__(256) void hgc_scale_rows(
    float4* __restrict__ d, const float* __restrict__ scale, int rows) {
  int j = blockIdx.x * blockDim.x + threadIdx.x;      // float4 index
  if (j >= rows * 32) return;
  float s = scale[j >> 5];
  float4 v = d[j];
  v.x *= s; v.y *= s; v.z *= s; v.w *= s;
  d[j] = v;
}

// ---------------- epilogue: out = elu(acc * dinv[row] + bias) --------------
__global__ __launch_bounds__(256) void hgc_finalize(
    const float4* __restrict__ acc, const float* __restrict__ dinv,
    const float4* __restrict__ bias, float4* __restrict__ out, int rows) {
  int j = blockIdx.x * blockDim.x + threadIdx.x;      // float4 index
  if (j >= rows * 32) return;
  float s  = dinv[j >> 5];
  float4 b = bias[j & 31];
  float4 v = acc[j];
  float t;
  t = v.x * s + b.x; v.x = (t > 0.f) ? t : expm1f(t);
  t = v.y * s + b.y; v.y = (t > 0.f) ? t : expm1f(t);
  t = v.z * s + b.z; v.z = (t > 0.f) ? t : expm1f(t);
  t = v.w * s + b.w; v.w = (t > 0.f) ? t : expm1f(t);
  out[j] = v;
}

// ---------------------------------------------------------------------------
extern "C" void kernel_launch(void* const* d_in, const int* in_sizes, int n_in,
                              void* d_out, int out_size, void* d_ws, size_t ws_size,
                              hipStream_t stream) {
  const float* x   = (const float*)d_in[0];
  const float* W1  = (const float*)d_in[1];
  const float* b1  = (const float*)d_in[2];
  const float* W2  = (const float*)d_in[3];
  const float* b2  = (const float*)d_in[4];
  const float* W3  = (const float*)d_in[5];
  const float* b3  = (const float*)d_in[6];
  const int* node_idx  = (const int*)d_in[7];
  const int* hedge_idx = (const int*)d_in[8];

  const int N = in_sizes[0] / 128;
  const int E = in_sizes[7];
  const int M = 25000;  // num_hyperedges: device scalar d_in[9] unreadable during
                        // graph capture; fixed by the problem spec.

  float* ws     = (float*)d_ws;
  float* XW     = ws;  ws += (size_t)N * 128;   // GEMM output
  float* accum  = ws;  ws += (size_t)N * 128;   // layer ping buffer
  float* medge  = ws;  ws += (size_t)M * 128;   // hyperedge features
  float* dn     = ws;  ws += (size_t)N;         // D^-1
  float* de     = ws;  ws += (size_t)M;         // B^-1

  // ---- degrees once (indices constant across layers) ----
  hipMemsetAsync(dn, 0, (size_t)N * sizeof(float), stream);
  hipMemsetAsync(de, 0, (size_t)M * sizeof(float), stream);
  hgc_degrees<<<(E + 255) / 256, 256, 0, stream>>>(node_idx, hedge_idx, dn, de, E);
  hgc_invert<<<(N + 255) / 256, 256, 0, stream>>>(dn, N);
  hgc_invert<<<(M + 255) / 256, 256, 0, stream>>>(de, M);

  auto layer = [&](const float* in, const float* W, const float* b, float* out) {
    hgc_gemm_wmma<<<(N + 15) / 16, 256, 0, stream>>>(in, W, XW, N);
    hipMemsetAsync(medge, 0, (size_t)M * 128 * sizeof(float), stream);
    hgc_scatter_add<<<(E + 7) / 8, 256, 0, stream>>>(
        (const float4*)XW, node_idx, hedge_idx, medge, E);
    hgc_scale_rows<<<(M * 32 + 255) / 256, 256, 0, stream>>>(
        (float4*)medge, de, M);
    hipMemsetAsync(out, 0, (size_t)N * 128 * sizeof(float), stream);
    hgc_scatter_add<<<(E + 7) / 8, 256, 0, stream>>>(
        (const float4*)medge, hedge_idx, node_idx, out, E);
    hgc_finalize<<<(N * 32 + 255) / 256, 256, 0, stream>>>(
        (const float4*)out, dn, (const float4*)b, (float4*)out, N);
  };

  float* out_f = (float*)d_out;
  layer(x,     W1, b1, accum);   // layer 1: x -> accum
  layer(accum, W2, b2, accum);   // layer 2: accum -> accum (GEMM drains it first)
  layer(accum, W3, b3, out_f);   // layer 3: accum -> d_out
}